// CFRNet_Loss_10161892622626
// MI455X (gfx1250) — compile-verified
//
#include <hip/hip_runtime.h>
#include <hip/hip_bf16.h>
#include <math.h>

// ---------------------------------------------------------------------------
// CFRNet loss on MI455X (gfx1250, wave32).
//
// Pipeline (all on `stream`, fully device-side dynamic sizing):
//  1) partition : 1 wave, ballot-scan compaction of treatment==1/0 indices,
//                 masked MSE partial sums, p / marginal scalars.
//  2) gather    : Xt = out[idx1], Xc = out[idx0]  (2 MB each, L2 resident)
//  3) norms     : row squared-norms nx, ny
//  4) init_u    : u = a  (Sinkhorn init)
//  5) gemm_m    : WMMA f32 16x16x4 GEMM tiles -> M = sqrt(max(nx+ny-2*XtXc^T,eps))
//                 + fused mean/max reduction (atomics, positive-float uint max)
//  6) prep      : eff_lam = LAM/mean(M), delta = max(M), kd = exp(-lam*delta)+1e-6
//  7) m2e       : M -> E = exp(-lam*M) in place (pay 64M exps ONCE, not 21x)
//  8) 10x Sinkhorn: colpass (w = b / K^T u), rowpass (u = a / K w), K = E+1e-6,
//                 augmented row/col handled analytically via kd.
//  9) colpass   : v = b / K^T u (final)
// 10) vecsum    : Usum, Vsum (deterministic block reduce)
// 11) final_main: D_core = sum u_i v_j (E+1e-6) * (-log(E)/lam)   (recover M)
// 12) finalize  : scalar assembly -> d_out[0]
//
// Roofline: ~22 passes over the 256MB E/M matrix ~ 5.6 GB -> ~0.24 ms @ 23.3 TB/s.
// GEMM is 8.6 GFLOP fp32 -> negligible vs memory; hence f32 WMMA (no bf16).
// ---------------------------------------------------------------------------

typedef float v2f __attribute__((ext_vector_type(2)));
typedef float v8f __attribute__((ext_vector_type(8)));

#define LAM_C   10.0f
#define EPS_C   1e-6f

// header slots (floats unless noted)
// 0:nt(int) 1:nc(int) 2:sumsq0 3:sumsq1 4:Msum 5:Mmax(uint bits) 6:eff_lam
// 7:delta 8:kd 9:p 10:a0 11:aN 12:b0 13:bN 14:Dacc 15:Usum 16:Vsum

__global__ void init_hdr_k(float* wsf) {
  if (threadIdx.x < 64) wsf[threadIdx.x] = 0.0f;
}

__global__ void partition_k(const float* __restrict__ log_h,
                            const float* __restrict__ dur,
                            const int*   __restrict__ treat,
                            float* wsf, int* idx1, int* idx0, int n) {
  int lane = threadIdx.x;                       // single wave32
  int c1 = 0, c0 = 0; float s1 = 0.f, s0 = 0.f;
  for (int base = 0; base < n; base += 32) {
    int i = base + lane;
    int t = (i < n) ? treat[i] : -1;
    float d = (i < n) ? (log_h[i] - dur[i]) : 0.f;
    bool is1 = (t == 1), is0 = (t == 0);
    unsigned m1 = (unsigned)__ballot(is1);
    unsigned m0 = (unsigned)__ballot(is0);
    unsigned pre = (1u << lane) - 1u;
    if (is1) { idx1[c1 + __popc(m1 & pre)] = i; s1 += d * d; }
    if (is0) { idx0[c0 + __popc(m0 & pre)] = i; s0 += d * d; }
    c1 += __popc(m1); c0 += __popc(m0);
  }
  for (int o = 16; o > 0; o >>= 1) {
    s1 += __shfl_xor(s1, o, 32);
    s0 += __shfl_xor(s0, o, 32);
  }
  if (lane == 0) {
    ((int*)wsf)[0] = c1;  ((int*)wsf)[1] = c0;
    wsf[2] = s0;  wsf[3] = s1;
    float p = (float)c1 / (float)n;
    wsf[9]  = p;
    wsf[10] = p / (float)c1;            // a0 (== 1/n)
    wsf[11] = 1.f - p;                  // aN
    wsf[12] = (1.f - p) / (float)c0;    // b0 (== 1/n)
    wsf[13] = p;                        // bN
  }
}

__global__ void gather_k(const float* __restrict__ outm,
                         const int* __restrict__ idx1, const int* __restrict__ idx0,
                         const float* wsf, float* XT, float* XC) {
  int nt = ((const int*)wsf)[0], nc = ((const int*)wsf)[1];
  int tid = blockIdx.x * blockDim.x + threadIdx.x;
  int r = tid >> 6, c = tid & 63;
  if (r < nt) XT[(size_t)r * 64 + c] = outm[(size_t)idx1[r] * 64 + c];
  if (r < nc) XC[(size_t)r * 64 + c] = outm[(size_t)idx0[r] * 64 + c];
}

__global__ void norms_k(const float* wsf, const float* __restrict__ XT,
                        const float* __restrict__ XC, float* NX, float* NY) {
  int nt = ((const int*)wsf)[0], nc = ((const int*)wsf)[1];
  int r = blockIdx.x * blockDim.x + threadIdx.x;
  if (r < nt) {
    float s = 0.f;
    #pragma unroll 8
    for (int c = 0; c < 64; ++c) { float v = XT[(size_t)r * 64 + c]; s += v * v; }
    NX[r] = s;
  }
  if (r < nc) {
    float s = 0.f;
    #pragma unroll 8
    for (int c = 0; c < 64; ++c) { float v = XC[(size_t)r * 64 + c]; s += v * v; }
    NY[r] = s;
  }
}

__global__ void init_u_k(const float* wsf, float* U) {
  int nt = ((const int*)wsf)[0];
  float a0 = wsf[10], aN = wsf[11];
  int i = blockIdx.x * blockDim.x + threadIdx.x;
  if (i < nt) U[i] = a0; else if (i == nt) U[i] = aN;
}

// ---- WMMA GEMM: block = 8 waves, tile 32(rows) x 64(cols), LDS-staged ------
__global__ __launch_bounds__(256) void gemm_m_k(float* wsf,
                                                const float* __restrict__ XT,
                                                const float* __restrict__ XC,
                                                const float* __restrict__ NX,
                                                const float* __restrict__ NY,
                                                float* __restrict__ Mb) {
  int nt = ((int*)wsf)[0], nc = ((int*)wsf)[1];
  int ncp = (nc + 63) & ~63;
  int ti = blockIdx.y * 32, tj = blockIdx.x * 64;
  if (ti >= nt || tj >= nc) return;

  __shared__ float As[32 * 64];   // Xt tile  [row][k]
  __shared__ float Bs[64 * 64];   // Xc tile  [col][k]
  for (int k = threadIdx.x; k < 32 * 64; k += 256) {
    int r = k >> 6, c = k & 63;
    As[k] = (ti + r < nt) ? XT[(size_t)(ti + r) * 64 + c] : 0.f;
  }
  for (int k = threadIdx.x; k < 64 * 64; k += 256) {
    int r = k >> 6, c = k & 63;
    Bs[k] = (tj + r < nc) ? XC[(size_t)(tj + r) * 64 + c] : 0.f;
  }
  __syncthreads();

  int wave = threadIdx.x >> 5, lane = threadIdx.x & 31;
  int wr = (wave >> 2) * 16;          // wave row offset in tile (0 or 16)
  int wc = (wave & 3) * 16;           // wave col offset in tile (0..48)
  int half = lane >> 4, l16 = lane & 15;

  // f32 A 16x4 layout: lanes 0-15 hold K=k0..k0+1, lanes 16-31 K=k0+2..k0+3
  const float* Ap = &As[(size_t)(wr + l16) * 64 + 2 * half];
  const float* Bp = &Bs[(size_t)(wc + l16) * 64 + 2 * half];

  v8f acc = {0.f, 0.f, 0.f, 0.f, 0.f, 0.f, 0.f, 0.f};
  #pragma unroll
  for (int k0 = 0; k0 < 64; k0 += 4) {
    v2f av = *(const v2f*)(Ap + k0);            // ds_load_b64
    v2f bv = *(const v2f*)(Bp + k0);
    acc = __builtin_amdgcn_wmma_f32_16x16x4_f32(
        /*neg_a=*/false, av, /*neg_b=*/false, bv,
        /*c_mod=*/(short)0, acc, /*reuse_a=*/false, /*reuse_b=*/false);
  }

  // epilogue: M = sqrt(max(nx+ny-2*dot, eps)); fused sum/max reduction
  float lsum = 0.f, lmax = 0.f;
  #pragma unroll
  for (int r = 0; r < 8; ++r) {
    int gi = ti + wr + r + 8 * half;            // C/D layout: lanes 16-31 = M+8
    int gj = tj + wc + l16;
    if (gi < nt && gj < nc) {
      float m = sqrtf(fmaxf(NX[gi] + NY[gj] - 2.f * acc[r], EPS_C));
      Mb[(size_t)gi * ncp + gj] = m;
      lsum += m; lmax = fmaxf(lmax, m);
    }
  }
  for (int o = 16; o > 0; o >>= 1) {
    lsum += __shfl_xor(lsum, o, 32);
    lmax = fmaxf(lmax, __shfl_xor(lmax, o, 32));
  }
  if (lane == 0) {
    atomicAdd(&wsf[4], lsum);
    atomicMax((unsigned*)&wsf[5], __float_as_uint(lmax));  // positive floats
  }
}

__global__ void prep_k(float* wsf) {
  int nt = ((int*)wsf)[0], nc = ((int*)wsf)[1];
  float mean  = wsf[4] / ((float)nt * (float)nc);
  float el    = LAM_C / mean;
  float delta = __uint_as_float(((unsigned*)wsf)[5]);
  wsf[6] = el;
  wsf[7] = delta;
  wsf[8] = __expf(-el * delta) + 1e-6f;          // kd
}

// M -> E = exp(-lam*M), in place; one row per block
__global__ void m2e_k(const float* wsf, float* Mb) {
  int nt = ((const int*)wsf)[0], nc = ((const int*)wsf)[1];
  int ncp = (nc + 63) & ~63;
  float el = wsf[6];
  int i = blockIdx.x;
  if (i >= nt) return;
  float* Mr = Mb + (size_t)i * ncp;
  for (int j = threadIdx.x; j < nc; j += blockDim.x)
    Mr[j] = __expf(-el * Mr[j]);
}

// w = b / (K^T u); thread per column, coalesced across lanes
__global__ __launch_bounds__(256) void colpass_k(const float* wsf,
                                                 const float* __restrict__ Eb,
                                                 const float* __restrict__ U,
                                                 float* __restrict__ W) {
  int nt = ((const int*)wsf)[0], nc = ((const int*)wsf)[1];
  int ncp = (nc + 63) & ~63;
  float kd = wsf[8], b0 = wsf[12], bN = wsf[13];
  int j = blockIdx.x * blockDim.x + threadIdx.x;
  if (j > nc) return;
  float acc = 0.f;
  if (j < nc) {
    const float* Ec = Eb + j;
    for (int i = 0; i < nt; ++i)
      acc = fmaf(Ec[(size_t)i * ncp] + 1e-6f, U[i], acc);
    acc += kd * U[nt];
    W[j] = b0 / acc;
  } else {                                       // augmented column (Mt = delta)
    float us = 0.f;
    for (int i = 0; i < nt; ++i) us += U[i];
    acc = kd * us + (1.f + 1e-6f) * U[nt];
    W[nc] = bN / acc;
  }
}

// u = a / (K w); wave per row, shuffle reduce
__global__ __launch_bounds__(256) void rowpass_k(const float* wsf,
                                                 const float* __restrict__ Eb,
                                                 const float* __restrict__ W,
                                                 float* __restrict__ U) {
  int nt = ((const int*)wsf)[0], nc = ((const int*)wsf)[1];
  int ncp = (nc + 63) & ~63;
  float kd = wsf[8], a0 = wsf[10], aN = wsf[11];
  int row  = blockIdx.x * 8 + (threadIdx.x >> 5);
  int lane = threadIdx.x & 31;
  if (row > nt) return;
  float acc = 0.f;
  if (row < nt) {
    const float* Er = Eb + (size_t)row * ncp;
    for (int j = lane; j < nc; j += 32)
      acc = fmaf(Er[j] + 1e-6f, W[j], acc);
  } else {
    for (int j = lane; j < nc; j += 32) acc += W[j];
  }
  for (int o = 16; o > 0; o >>= 1) acc += __shfl_xor(acc, o, 32);
  if (lane == 0) {
    if (row < nt) U[row] = a0 / (acc + kd * W[nc]);
    else          U[nt]  = aN / (kd * acc + (1.f + 1e-6f) * W[nc]);
  }
}

__global__ void vecsum_k(float* wsf, const float* __restrict__ U,
                         const float* __restrict__ W) {
  __shared__ float su[256], sv[256];
  int nt = ((const int*)wsf)[0], nc = ((const int*)wsf)[1];
  float us = 0.f, vs = 0.f;
  for (int i = threadIdx.x; i < nt; i += 256) us += U[i];
  for (int j = threadIdx.x; j < nc; j += 256) vs += W[j];
  su[threadIdx.x] = us; sv[threadIdx.x] = vs;
  __syncthreads();
  for (int o = 128; o > 0; o >>= 1) {
    if ((int)threadIdx.x < o) {
      su[threadIdx.x] += su[threadIdx.x + o];
      sv[threadIdx.x] += sv[threadIdx.x + o];
    }
    __syncthreads();
  }
  if (threadIdx.x == 0) { wsf[15] = su[0]; wsf[16] = sv[0]; }
}

// D_core = sum_{i<nt,j<nc} u_i v_j (E+1e-6) * M,  M recovered as -log(E)/lam
__global__ __launch_bounds__(256) void final_main_k(float* wsf,
                                                    const float* __restrict__ Eb,
                                                    const float* __restrict__ U,
                                                    const float* __restrict__ W) {
  int nt = ((const int*)wsf)[0], nc = ((const int*)wsf)[1];
  int ncp = (nc + 63) & ~63;
  float iel = 1.f / wsf[6];
  int row  = blockIdx.x * 8 + (threadIdx.x >> 5);
  int lane = threadIdx.x & 31;
  if (row >= nt) return;
  const float* Er = Eb + (size_t)row * ncp;
  float acc = 0.f;
  for (int j = lane; j < nc; j += 32) {
    float e = Er[j];
    float m = -__logf(fmaxf(e, 1e-30f)) * iel;
    acc = fmaf((e + 1e-6f) * W[j], m, acc);
  }
  for (int o = 16; o > 0; o >>= 1) acc += __shfl_xor(acc, o, 32);
  if (lane == 0) atomicAdd(&wsf[14], U[row] * acc);
}

__global__ void finalize_k(const float* wsf, const float* U, const float* W,
                           float* outp) {
  int nt = ((const int*)wsf)[0], nc = ((const int*)wsf)[1];
  float p = wsf[9];
  float loss0 = wsf[2] / (float)nc;
  float loss1 = wsf[3] / (float)nt;
  float kd = wsf[8], delta = wsf[7];
  // augmented row/col contributions are rank-1: kd*delta*(u_N*Vsum + Usum*v_N)
  float D = 2.f * (wsf[14] + kd * delta * (U[nt] * wsf[16] + wsf[15] * W[nc]));
  outp[0] = (1.f - p) * loss0 + p * loss1 + 1.0f /*ALPHA*/ * D;
}

extern "C" void kernel_launch(void* const* d_in, const int* in_sizes, int n_in,
                              void* d_out, int out_size, void* d_ws, size_t ws_size,
                              hipStream_t stream) {
  const float* log_h = (const float*)d_in[0];
  const float* outm  = (const float*)d_in[1];
  const float* dur   = (const float*)d_in[2];
  const int*   treat = (const int*)d_in[3];
  float* res = (float*)d_out;
  const int N = in_sizes[0];                     // 16384; dim fixed at 64

  // workspace carve-up (floats)
  float* wsf = (float*)d_ws;
  float* U   = wsf + 256;                        // nt+1 (padded N+32)
  float* W   = U   + (N + 32);                   // nc+1
  float* NX  = W   + (N + 32);
  float* NY  = NX  + N;
  float* XT  = NY  + N;
  float* XC  = XT  + (size_t)N * 64;
  int*   I1  = (int*)(XC + (size_t)N * 64);
  int*   I0  = I1 + N;
  float* Mb  = (float*)(I0 + N);                 // nt x ncp, up to ~268 MB

  init_hdr_k<<<1, 64, 0, stream>>>(wsf);
  partition_k<<<1, 32, 0, stream>>>(log_h, dur, treat, wsf, I1, I0, N);
  gather_k<<<(N * 64 + 255) / 256, 256, 0, stream>>>(outm, I1, I0, wsf, XT, XC);
  norms_k<<<(N + 255) / 256, 256, 0, stream>>>(wsf, XT, XC, NX, NY);
  init_u_k<<<(N + 1 + 255) / 256, 256, 0, stream>>>(wsf, U);

  dim3 gg((N + 63) / 64, (N + 31) / 32);         // worst-case, device-guarded
  gemm_m_k<<<gg, 256, 0, stream>>>(wsf, XT, XC, NX, NY, Mb);

  prep_k<<<1, 1, 0, stream>>>(wsf);
  m2e_k<<<N, 256, 0, stream>>>(wsf, Mb);

  const int colBlocks = (N + 1 + 255) / 256;
  const int rowBlocks = (N + 1 + 7) / 8;
  for (int it = 0; it < 10; ++it) {
    colpass_k<<<colBlocks, 256, 0, stream>>>(wsf, Mb, U, W);
    rowpass_k<<<rowBlocks, 256, 0, stream>>>(wsf, Mb, W, U);
  }
  colpass_k<<<colBlocks, 256, 0, stream>>>(wsf, Mb, U, W);   // v = W

  vecsum_k<<<1, 256, 0, stream>>>(wsf, U, W);
  final_main_k<<<(N + 7) / 8, 256, 0, stream>>>(wsf, Mb, U, W);
  finalize_k<<<1, 1, 0, stream>>>(wsf, U, W, res);
}